// TaskAlignedAssigner_10230612099530
// MI455X (gfx1250) — compile-verified
//
#include <hip/hip_runtime.h>

// Problem constants (fixed by setup_inputs: img=640, strides 8/16/32)
#define BS   32
#define NA   8400
#define NG   64
#define NC   80
#define KTOP 10
#define EPSF     1e-9f
#define IOU_EPSF 1e-7f
#define ATILE 128   // anchors per block in assign kernel (8 waves x 16)

typedef __attribute__((ext_vector_type(2))) float v2f;
typedef __attribute__((ext_vector_type(8))) float v8f;

__device__ __forceinline__ float iou_xyxy(float p0, float p1, float p2, float p3,
                                          float g0, float g1, float g2, float g3) {
  float iw = fmaxf(fminf(p2, g2) - fmaxf(p0, g0), 0.f);
  float ih = fmaxf(fminf(p3, g3) - fmaxf(p1, g1), 0.f);
  float inter = iw * ih;
  float a1 = (p2 - p0) * (p3 - p1);
  float a2 = (g2 - g0) * (g3 - g1);
  return inter / (a1 + a2 - inter + IOU_EPSF);
}

// ---------------------------------------------------------------------------
// Stage 1: per (b,g) compute the 10th-largest align metric over all anchors.
// One block per (g,b). align values staged in LDS (8400 floats = 33.6 KB,
// comfortably inside the 320KB WGP LDS), then 10 iterative max-reductions.
// ---------------------------------------------------------------------------
__global__ __launch_bounds__(256) void topk_thresh_kernel(
    const float* __restrict__ pd_scores, const float* __restrict__ pd_bboxes,
    const float* __restrict__ anc, const int* __restrict__ gt_labels,
    const float* __restrict__ gt_bboxes, const unsigned char* __restrict__ mask_gt,
    float* __restrict__ Tout) {
  __shared__ float sm[NA];
  __shared__ float rv[256];
  __shared__ int   ri[256];

  const int tid = threadIdx.x;
  const int g = blockIdx.x, b = blockIdx.y;

  const float gx0 = gt_bboxes[(size_t)((b * NG) + g) * 4 + 0];
  const float gy0 = gt_bboxes[(size_t)((b * NG) + g) * 4 + 1];
  const float gx1 = gt_bboxes[(size_t)((b * NG) + g) * 4 + 2];
  const float gy1 = gt_bboxes[(size_t)((b * NG) + g) * 4 + 3];
  const int   lab = gt_labels[b * NG + g];
  const int   valid = (mask_gt[b * NG + g] != 0) ? 1 : 0;

  const float* __restrict__ sc = pd_scores + (size_t)b * NA * NC + lab;
  const float* __restrict__ pb = pd_bboxes + (size_t)b * NA * 4;

  for (int a = tid; a < NA; a += 256) {
    float ax = anc[a * 2 + 0], ay = anc[a * 2 + 1];
    float d0 = ax - gx0, d1 = ay - gy0, d2 = gx1 - ax, d3 = gy1 - ay;
    float mn = fminf(fminf(d0, d1), fminf(d2, d3));
    float m = 0.f;
    if (valid && (mn > EPSF)) {
      float s  = sc[(size_t)a * NC];
      float p0 = pb[(size_t)a * 4 + 0], p1 = pb[(size_t)a * 4 + 1];
      float p2 = pb[(size_t)a * 4 + 2], p3 = pb[(size_t)a * 4 + 3];
      float iou = iou_xyxy(p0, p1, p2, p3, gx0, gy0, gx1, gy1);
      float i2 = iou * iou;
      m = sqrtf(s) * (i2 * i2 * i2);
    }
    sm[a] = m;
  }
  __syncthreads();

  float T = 0.f;
  for (int it = 0; it < KTOP; ++it) {
    float bv = -1.f; int bi = 0;
    for (int a = tid; a < NA; a += 256) {
      float v = sm[a];
      if (v > bv) { bv = v; bi = a; }
    }
    rv[tid] = bv; ri[tid] = bi;
    __syncthreads();
    for (int s = 128; s > 0; s >>= 1) {
      if (tid < s && rv[tid + s] > rv[tid]) { rv[tid] = rv[tid + s]; ri[tid] = ri[tid + s]; }
      __syncthreads();
    }
    T = rv[0];
    __syncthreads();
    if (tid == 0) sm[ri[0]] = -1.f;  // exclude this index from further rounds
    __syncthreads();
  }
  if (tid == 0) Tout[b * NG + g] = fmaxf(T, 0.f);
}

// ---------------------------------------------------------------------------
// Stage 2: main assignment. Each wave gathers scores for 16 anchors x 64 gts
// as a one-hot GEMM using V_WMMA_F32_16X16X4_F32 (exact in fp32), spills the
// tile to padded LDS, then one thread per anchor runs the g-loop.
// ---------------------------------------------------------------------------
__global__ __launch_bounds__(256) void assign_kernel(
    const float* __restrict__ pd_scores, const float* __restrict__ pd_bboxes,
    const float* __restrict__ anc, const int* __restrict__ gt_labels,
    const float* __restrict__ gt_bboxes, const unsigned char* __restrict__ mask_gt,
    const float* __restrict__ Tin, float* __restrict__ out) {
  __shared__ float sLDS[ATILE * 65];   // gathered scores, padded (stride 65)
  __shared__ float gbox[NG * 4];
  __shared__ int   glab[NG];
  __shared__ int   gval[NG];
  __shared__ float gT[NG];
  __shared__ int   albl[ATILE];
  __shared__ float atsc[ATILE];
  __shared__ int   afg[ATILE];

  const int tid = threadIdx.x;
  const int b = blockIdx.y;
  const int aBase = blockIdx.x * ATILE;

  if (tid < NG) {
    glab[tid] = gt_labels[b * NG + tid];
    gval[tid] = (mask_gt[b * NG + tid] != 0) ? 1 : 0;
    gT[tid]   = Tin[b * NG + tid];
  }
  if (tid < NG * 4) gbox[tid] = gt_bboxes[(size_t)(b * NG) * 4 + tid];
  __syncthreads();

  // ---- WMMA one-hot gather: gathered[a,g] = pd_scores[b,a,:] . onehot(lab[g])
  const int wave = tid >> 5, lane = tid & 31;
  const int row = lane & 15, half = lane >> 4;
  int aRow = aBase + wave * 16 + row;
  if (aRow > NA - 1) aRow = NA - 1;  // clamp: keeps EXEC all-ones for WMMA
  const float* __restrict__ srow = pd_scores + ((size_t)b * NA + aRow) * NC;
  const int l0 = glab[row], l1 = glab[16 + row], l2 = glab[32 + row], l3 = glab[48 + row];

  v8f acc0 = {0.f,0.f,0.f,0.f,0.f,0.f,0.f,0.f};
  v8f acc1 = acc0, acc2 = acc0, acc3 = acc0;

#pragma unroll
  for (int k = 0; k < NC / 4; ++k) {
    const int kb = k * 4 + 2 * half;       // K-offset this half-wave supplies
    v2f A = *(const v2f*)(srow + kb);      // A: 16x4 fp32 chunk (lane-striped)
    v2f B0, B1, B2, B3;                    // B: 4x16 one-hot chunk per g-tile
    B0.x = (l0 == kb) ? 1.f : 0.f;  B0.y = (l0 == kb + 1) ? 1.f : 0.f;
    B1.x = (l1 == kb) ? 1.f : 0.f;  B1.y = (l1 == kb + 1) ? 1.f : 0.f;
    B2.x = (l2 == kb) ? 1.f : 0.f;  B2.y = (l2 == kb + 1) ? 1.f : 0.f;
    B3.x = (l3 == kb) ? 1.f : 0.f;  B3.y = (l3 == kb + 1) ? 1.f : 0.f;
    acc0 = __builtin_amdgcn_wmma_f32_16x16x4_f32(false, A, false, B0, (short)0, acc0, false, false);
    acc1 = __builtin_amdgcn_wmma_f32_16x16x4_f32(false, A, false, B1, (short)0, acc1, false, false);
    acc2 = __builtin_amdgcn_wmma_f32_16x16x4_f32(false, A, false, B2, (short)0, acc2, false, false);
    acc3 = __builtin_amdgcn_wmma_f32_16x16x4_f32(false, A, false, B3, (short)0, acc3, false, false);
  }

  // C/D layout: VGPR j holds M=j (lanes 0-15) / M=j+8 (lanes 16-31), N=lane&15
  const int aLoc = wave * 16 + 8 * half;
#pragma unroll
  for (int j = 0; j < 8; ++j) {
    sLDS[(aLoc + j) * 65 + row]      = acc0[j];
    sLDS[(aLoc + j) * 65 + 16 + row] = acc1[j];
    sLDS[(aLoc + j) * 65 + 32 + row] = acc2[j];
    sLDS[(aLoc + j) * 65 + 48 + row] = acc3[j];
  }
  __syncthreads();

  const size_t O0 = 0;
  const size_t O1 = (size_t)BS * NA;
  const size_t O2 = O1 + (size_t)BS * NA * 4;
  const size_t O3 = O2 + (size_t)BS * NA * NC;
  const size_t O4 = O3 + (size_t)BS * NA;

  if (tid < ATILE) {
    const int a = aBase + tid;
    int lbl = 0, fg = 0, tgi = 0;
    float tsc = 0.f;
    if (a < NA) {
      const float ax = anc[a * 2 + 0], ay = anc[a * 2 + 1];
      const float* pr = pd_bboxes + ((size_t)b * NA + a) * 4;
      const float p0 = pr[0], p1 = pr[1], p2 = pr[2], p3 = pr[3];
      int npos = 0, first_pos = 0, best_g = 0;
      float best_ov = -1.f, mx_al = 0.f, mx_ov = 0.f, mx_prod = 0.f;
      for (int g = 0; g < NG; ++g) {
        const float g0 = gbox[g * 4 + 0], g1 = gbox[g * 4 + 1];
        const float g2 = gbox[g * 4 + 2], g3 = gbox[g * 4 + 3];
        float d0 = ax - g0, d1 = ay - g1, d2 = g2 - ax, d3 = g3 - ay;
        float mn = fminf(fminf(d0, d1), fminf(d2, d3));
        bool mask = gval[g] && (mn > EPSF);
        float iou = iou_xyxy(p0, p1, p2, p3, g0, g1, g2, g3);
        float ov = mask ? iou : 0.f;
        float s  = sLDS[tid * 65 + g];
        float i2 = iou * iou;
        float al = mask ? sqrtf(s) * (i2 * i2 * i2) : 0.f;
        if (ov > best_ov) { best_ov = ov; best_g = g; }  // first-max (argmax)
        if (al > 0.f && al >= gT[g]) { if (!npos) first_pos = g; ++npos; }
        mx_al   = fmaxf(mx_al, al);
        mx_ov   = fmaxf(mx_ov, ov);
        mx_prod = fmaxf(mx_prod, al * ov);
      }
      fg  = (npos > 0) ? 1 : 0;
      tgi = (npos > 1) ? best_g : (npos == 1 ? first_pos : 0);
      lbl = glab[tgi]; if (lbl < 0) lbl = 0;
      if (fg) tsc = mx_prod / (fmaxf(mx_al, EPSF) * fmaxf(mx_ov, EPSF));

      const size_t base = (size_t)b * NA + a;
      out[O0 + base] = (float)lbl;
      out[O1 + base * 4 + 0] = gbox[tgi * 4 + 0];
      out[O1 + base * 4 + 1] = gbox[tgi * 4 + 1];
      out[O1 + base * 4 + 2] = gbox[tgi * 4 + 2];
      out[O1 + base * 4 + 3] = gbox[tgi * 4 + 3];
      out[O3 + base] = fg ? 1.f : 0.f;
      out[O4 + base] = (float)tgi;
    }
    albl[tid] = lbl; atsc[tid] = tsc; afg[tid] = fg;
  }
  __syncthreads();

  // Cooperative, coalesced write of one-hot target_scores rows (80 floats/anchor)
  for (int idx = tid; idx < ATILE * NC; idx += 256) {
    const int al_ = idx / NC, c = idx % NC;
    const int a = aBase + al_;
    if (a < NA) {
      float v = (afg[al_] && c == albl[al_]) ? atsc[al_] : 0.f;
      out[O2 + ((size_t)b * NA + a) * NC + c] = v;
    }
  }
}

extern "C" void kernel_launch(void* const* d_in, const int* in_sizes, int n_in,
                              void* d_out, int out_size, void* d_ws, size_t ws_size,
                              hipStream_t stream) {
  (void)in_sizes; (void)n_in; (void)out_size; (void)ws_size;
  const float*         pd_scores = (const float*)d_in[0];
  const float*         pd_bboxes = (const float*)d_in[1];
  const float*         anc       = (const float*)d_in[2];
  const int*           gt_labels = (const int*)d_in[3];
  const float*         gt_bboxes = (const float*)d_in[4];
  const unsigned char* mask_gt   = (const unsigned char*)d_in[5];  // jnp bool -> 1 byte
  float* out = (float*)d_out;
  float* Tws = (float*)d_ws;  // BS*NG floats = 8 KB of scratch

  dim3 g1(NG, BS);
  topk_thresh_kernel<<<g1, 256, 0, stream>>>(pd_scores, pd_bboxes, anc,
                                             gt_labels, gt_bboxes, mask_gt, Tws);
  dim3 g2((NA + ATILE - 1) / ATILE, BS);
  assign_kernel<<<g2, 256, 0, stream>>>(pd_scores, pd_bboxes, anc,
                                        gt_labels, gt_bboxes, mask_gt, Tws, out);
}